// SyntheticModelTFDE_15745350107764
// MI455X (gfx1250) — compile-verified
//
#include <hip/hip_runtime.h>
#include <hip/hip_bf16.h>

// Problem constants (from reference)
#define TT   26
#define RR   100000
#define DD   128
#define NNZ  10
#define BB   16384
#define NUMF 13
#define KIN  (TT*DD + NUMF)   // 3341 real input width
#define K0P  3392             // padded input width = 53 * 64
#define H0   1024
#define H1   512
#define H2   256
#define KC   64               // K chunk per TDM stage

// LDS row strides produced by TDM pad-on-store
#define A_STRIDE 67           // 64 + 3  (pad_amount code 2)
#define B_STRIDE 80           // 64 + 16 (pad_amount code 15)

typedef __attribute__((ext_vector_type(2))) float        v2f;
typedef __attribute__((ext_vector_type(8))) float        v8f;
typedef __attribute__((ext_vector_type(4))) unsigned int v4u;
typedef __attribute__((ext_vector_type(8))) int          v8i;
typedef __attribute__((ext_vector_type(4))) int          v4i;

// ---------------------------------------------------------------------------
// TDM 2-D tile load: global (row-major, stride_elems fp32/row) -> LDS,
// with pad_amount dwords appended after every 64 dwords stored.
// D# layout per CDNA5 ISA.
// 6-arg builtin form (clang-23 / therock-10.0 headers).
// ---------------------------------------------------------------------------
__device__ __forceinline__ void tdm_load_2d(unsigned lds_byte_addr,
                                            const float* gptr,
                                            unsigned tile_w,      // elements, contiguous
                                            unsigned tile_h,      // rows
                                            unsigned long long stride_elems,
                                            unsigned pad_amount_code) {
  unsigned long long ga = (unsigned long long)(size_t)gptr;
  v4u g0;
  g0[0] = 1u;                                             // count=1, user descriptor
  g0[1] = lds_byte_addr;                                  // lds_addr
  g0[2] = (unsigned)(ga & 0xFFFFFFFFu);                   // global_addr[31:0]
  g0[3] = (unsigned)((ga >> 32) & 0x01FFFFFFu) | (2u << 30);  // addr[56:32] | type=2
  v8i g1;
  // data_size=4B(2) | pad_enable | pad_interval=64dw(5) | pad_amount code
  g1[0] = (int)((2u << 16) | (1u << 20) | (5u << 22) | (pad_amount_code << 25));
  g1[1] = (int)((tile_w & 0xFFFFu) << 16);                // tensor_dim0[15:0]
  g1[2] = (int)(((tile_w >> 16) & 0xFFFFu) | ((tile_h & 0xFFFFu) << 16)); // dim0 hi | dim1 lo
  g1[3] = (int)(((tile_h >> 16) & 0xFFFFu) | ((tile_w & 0xFFFFu) << 16)); // dim1 hi | tile_dim0
  g1[4] = (int)(tile_h & 0xFFFFu);                        // tile_dim1 | tile_dim2=0
  g1[5] = (int)(stride_elems & 0xFFFFFFFFu);              // tensor_dim0_stride lo
  g1[6] = (int)((stride_elems >> 32) & 0xFFFFu);          // stride hi | dim1_stride=0
  g1[7] = 0;
  v4i z4; z4[0] = 0; z4[1] = 0; z4[2] = 0; z4[3] = 0;
  v8i z8; z8[0] = 0; z8[1] = 0; z8[2] = 0; z8[3] = 0;
  z8[4] = 0; z8[5] = 0; z8[6] = 0; z8[7] = 0;
  __builtin_amdgcn_tensor_load_to_lds(g0, g1, z4, z4, z8, 0);
}

// ---------------------------------------------------------------------------
// 1) Zero-pad W0 [3341,1024] -> W0p [3392,1024]
// ---------------------------------------------------------------------------
__global__ __launch_bounds__(256) void pad_w0_kernel(const float* __restrict__ W0,
                                                     float* __restrict__ W0p) {
  size_t i = (size_t)blockIdx.x * 256 + threadIdx.x;
  if (i >= (size_t)K0P * H0) return;
  size_t k = i / H0;
  W0p[i] = (k < (size_t)KIN) ? W0[i] : 0.0f;
}

// ---------------------------------------------------------------------------
// 2) Embedding bag: one wave per (b,t); 32 lanes x float4 = 128 = D.
// ---------------------------------------------------------------------------
__global__ __launch_bounds__(256) void embed_bag_kernel(const float* __restrict__ tables,
                                                        const int* __restrict__ cat,
                                                        float* __restrict__ X0) {
  const int gwave = (int)(((size_t)blockIdx.x * blockDim.x + threadIdx.x) >> 5);
  const int lane  = threadIdx.x & 31;
  if (gwave >= BB * TT) return;
  const int b = gwave / TT;
  const int t = gwave - b * TT;

  const int*   idx = cat + ((size_t)b * TT + t) * NNZ;
  const float* tab = tables + (size_t)t * RR * DD;

  float ax = 0.f, ay = 0.f, az = 0.f, aw = 0.f;
#pragma unroll
  for (int j = 0; j < NNZ; ++j) {
    const int r = idx[j];
    if (j + 1 < NNZ) {
      __builtin_prefetch(tab + (size_t)idx[j + 1] * DD, 0, 0);  // global_prefetch_b8
    }
    const float4* row = (const float4*)(tab + (size_t)r * DD);
    float4 v = row[lane];
    ax += v.x; ay += v.y; az += v.z; aw += v.w;
  }
  float4* out = (float4*)(X0 + (size_t)b * K0P + (size_t)t * DD);
  float4 acc; acc.x = ax; acc.y = ay; acc.z = az; acc.w = aw;
  out[lane] = acc;
}

// ---------------------------------------------------------------------------
// 3) Numerical features + zero pad into X0[:, 3328:3392]
// ---------------------------------------------------------------------------
__global__ __launch_bounds__(64) void pack_tail_kernel(const float* __restrict__ numerical,
                                                       float* __restrict__ X0) {
  const int b = blockIdx.x;
  const int i = threadIdx.x;                 // 0..63
  float v = (i < NUMF) ? numerical[(size_t)b * NUMF + i] : 0.0f;
  X0[(size_t)b * K0P + TT * DD + i] = v;
}

// ---------------------------------------------------------------------------
// 4) Fused GEMM + bias + ReLU, V_WMMA_F32_16X16X4_F32 fed from LDS tiles
//    staged by the Tensor Data Mover (double-buffered, TENSORcnt tracked).
//
//    Block = 256 threads (8 waves), block tile = 128(M) x 64(N).
//    Per K-chunk (KC=64): TDM loads A tile 128xKC (row stride 67 dw in LDS)
//    and B tile KCx64 (row stride 80 dw) — one DMA pair per block, shared by
//    all 8 waves. Each wave: 16 k-steps x 4 N-subtiles = 64 WMMAs per chunk.
// ---------------------------------------------------------------------------
__global__ __launch_bounds__(256) void gemm_bias_relu_wmma(const float* __restrict__ A,
                                                           const float* __restrict__ Bm,
                                                           const float* __restrict__ bias,
                                                           float* __restrict__ C,
                                                           int K, int Nn, int do_relu) {
  __shared__ float sA[2][128 * A_STRIDE];   // 2 x 34,304 B
  __shared__ float sB[2][KC * B_STRIDE];    // 2 x 20,480 B

  const int lane = threadIdx.x & 31;
  const int wv   = threadIdx.x >> 5;            // 0..7
  const int m0   = blockIdx.y * 128;            // block M origin
  const int n0   = blockIdx.x * 64;             // block N origin
  const int half = lane >> 4;                   // 0: K+0/1 ; 1: K+2/3
  const int l16  = lane & 15;

  v8f c[4];
#pragma unroll
  for (int nt = 0; nt < 4; ++nt)
#pragma unroll
    for (int i = 0; i < 8; ++i) c[nt][i] = 0.0f;

  const int nch = K / KC;

  // Prologue: stage chunk 0 into buffer 0 (wave 0 only; TDM ignores EXEC).
  if (wv == 0) {
    tdm_load_2d((unsigned)(size_t)&sA[0][0], A + (size_t)m0 * K, KC, 128,
                (unsigned long long)K, 2u);
    tdm_load_2d((unsigned)(size_t)&sB[0][0], Bm + n0, 64, KC,
                (unsigned long long)Nn, 15u);
  }

  for (int kc = 0; kc < nch; ++kc) {
    const int buf = kc & 1;
    if (wv == 0) {
      if (kc + 1 < nch) {
        const int kn = (kc + 1) * KC;
        tdm_load_2d((unsigned)(size_t)&sA[buf ^ 1][0],
                    A + (size_t)m0 * K + kn, KC, 128, (unsigned long long)K, 2u);
        tdm_load_2d((unsigned)(size_t)&sB[buf ^ 1][0],
                    Bm + (size_t)kn * Nn + n0, 64, KC, (unsigned long long)Nn, 15u);
        __builtin_amdgcn_s_wait_tensorcnt(2);   // current chunk's 2 loads done
      } else {
        __builtin_amdgcn_s_wait_tensorcnt(0);
      }
    }
    __syncthreads();

    const float* __restrict__ sAb = &sA[buf][0];
    const float* __restrict__ sBb = &sB[buf][0];
    const int arow = (wv * 16 + l16) * A_STRIDE + half * 2;
#pragma unroll 4
    for (int kk = 0; kk < KC; kk += 4) {
      v2f a;
      a.x = sAb[arow + kk];
      a.y = sAb[arow + kk + 1];
      const int brow = (kk + half * 2) * B_STRIDE + l16;
#pragma unroll
      for (int nt = 0; nt < 4; ++nt) {
        v2f bf;
        bf.x = sBb[brow + nt * 16];
        bf.y = sBb[brow + nt * 16 + B_STRIDE];
        c[nt] = __builtin_amdgcn_wmma_f32_16x16x4_f32(
            false, a, false, bf, (short)0, c[nt], false, false);
      }
    }
    __syncthreads();   // all waves done reading before buffer reuse
  }

  // Epilogue: bias + ReLU, scatter per C/D layout (VGPR i -> M = i + 8*half)
#pragma unroll
  for (int nt = 0; nt < 4; ++nt) {
    const int   n  = n0 + nt * 16 + l16;
    const float bv = bias[n];
#pragma unroll
    for (int i = 0; i < 8; ++i) {
      float v = c[nt][i] + bv;
      if (do_relu) v = fmaxf(v, 0.0f);
      C[(size_t)(m0 + wv * 16 + half * 8 + i) * Nn + n] = v;
    }
  }
}

// ---------------------------------------------------------------------------
// 5) Final projection [B,256] @ [256,1] + b3 : wave-per-row dot product.
// ---------------------------------------------------------------------------
__global__ __launch_bounds__(256) void final_proj_kernel(const float* __restrict__ X3,
                                                         const float* __restrict__ W3,
                                                         const float* __restrict__ b3,
                                                         float* __restrict__ out) {
  const int row  = (int)(((size_t)blockIdx.x * blockDim.x + threadIdx.x) >> 5);
  const int lane = threadIdx.x & 31;
  if (row >= BB) return;

  const float4* x = (const float4*)(X3 + (size_t)row * H2);
  const float4* w = (const float4*)W3;
  float acc = 0.0f;
#pragma unroll
  for (int j = 0; j < 2; ++j) {
    float4 xv = x[j * 32 + lane];
    float4 wv = w[j * 32 + lane];
    acc += xv.x * wv.x + xv.y * wv.y + xv.z * wv.z + xv.w * wv.w;
  }
#pragma unroll
  for (int off = 16; off > 0; off >>= 1) acc += __shfl_xor(acc, off, 32);
  if (lane == 0) out[row] = acc + b3[0];
}

// ---------------------------------------------------------------------------
// Launch
// ---------------------------------------------------------------------------
extern "C" void kernel_launch(void* const* d_in, const int* in_sizes, int n_in,
                              void* d_out, int out_size, void* d_ws, size_t ws_size,
                              hipStream_t stream) {
  const float* numerical = (const float*)d_in[0];
  const int*   cat       = (const int*)d_in[1];
  const float* tables    = (const float*)d_in[2];
  const float* W0 = (const float*)d_in[3];  const float* b0 = (const float*)d_in[4];
  const float* W1 = (const float*)d_in[5];  const float* b1 = (const float*)d_in[6];
  const float* W2 = (const float*)d_in[7];  const float* b2 = (const float*)d_in[8];
  const float* W3 = (const float*)d_in[9];  const float* b3 = (const float*)d_in[10];
  float* out = (float*)d_out;

  // Workspace layout (floats): X0 | W0p | X1 | X2 | X3  (~353 MB total)
  float* ws  = (float*)d_ws;
  size_t o   = 0;
  float* X0  = ws + o;  o += (size_t)BB * K0P;
  float* W0p = ws + o;  o += (size_t)K0P * H0;
  float* X1  = ws + o;  o += (size_t)BB * H0;
  float* X2  = ws + o;  o += (size_t)BB * H1;
  float* X3  = ws + o;  o += (size_t)BB * H2;
  (void)in_sizes; (void)n_in; (void)out_size; (void)ws_size;

  {
    size_t tot = (size_t)K0P * H0;
    pad_w0_kernel<<<dim3((unsigned)((tot + 255) / 256)), dim3(256), 0, stream>>>(W0, W0p);
  }
  embed_bag_kernel<<<dim3((BB * TT) / 8), dim3(256), 0, stream>>>(tables, cat, X0);
  pack_tail_kernel<<<dim3(BB), dim3(64), 0, stream>>>(numerical, X0);

  gemm_bias_relu_wmma<<<dim3(H0 / 64, BB / 128), dim3(256), 0, stream>>>(X0, W0p, b0, X1, K0P, H0, 1);
  gemm_bias_relu_wmma<<<dim3(H1 / 64, BB / 128), dim3(256), 0, stream>>>(X1, W1, b1, X2, H0, H1, 1);
  gemm_bias_relu_wmma<<<dim3(H2 / 64, BB / 128), dim3(256), 0, stream>>>(X2, W2, b2, X3, H1, H2, 1);

  final_proj_kernel<<<dim3(BB / 8), dim3(256), 0, stream>>>(X3, W3, b3, out);
}